// MTGNN_6691559047592
// MI455X (gfx1250) — compile-verified
//
#include <hip/hip_runtime.h>

// ---------------- types / helpers ----------------
typedef __attribute__((ext_vector_type(2))) float v2f;
typedef __attribute__((ext_vector_type(8))) float v8f;

#if defined(__HIP_DEVICE_COMPILE__) && __has_builtin(__builtin_amdgcn_wmma_f32_16x16x4_f32)
#define USE_WMMA 1
#else
#define USE_WMMA 0
#endif

__device__ __forceinline__ float warp_sum32(float v) {
#pragma unroll
  for (int o = 16; o > 0; o >>= 1) v += __shfl_xor(v, o, 32);
  return v;
}

__device__ __forceinline__ v8f wmma4(v2f a, v2f b, v8f c) {
#if USE_WMMA
  return __builtin_amdgcn_wmma_f32_16x16x4_f32(false, a, false, b, (short)0, c, false, false);
#else
  c[0] += a.x * b.x + a.y * b.y;  // placeholder; host pass only, real values from fallback
  return c;
#endif
}

// =====================================================================
// WMMA f32 GEMM:  Y[M,N] = act( X[M,K] @ W[K,N] + bias[N] + add[M,N] )
// act(v) = v>0 ? v : slope*v   (slope=1.0 -> identity)
// One wave computes a 32x64 tile: 8 accumulators (2 M-halves x 4 N-tiles),
// two A fragments share four B fragments -> 12 loads per 8 wmma.
// A frag (16x4 f32): lane l: M=l&15, K-pair = (l>>4)*2 -> {a.x,a.y}
// B frag (4x16 f32): lane l: N=l&15, rows kh*2 (b.x) and kh*2+1 (b.y)
// C/D (16x16 f32):   VGPR r: M = r + 8*(l>>4), N = l&15
// Fast path: unconditional loads + pointer stepping (no per-iter multiplies).
// Edge/tail paths: clamped addresses + select (v_cndmask), branch-free.
// =====================================================================
__global__ void __launch_bounds__(32)
k_gemm(const float* __restrict__ X, int ldx,
       const float* __restrict__ W, int ldw,
       const float* __restrict__ bias,
       const float* add, int ldadd,
       float* Y, int ldy,
       int M, int N, int K, float slope) {
  const int m0 = blockIdx.x << 5;
  const int n0 = blockIdx.y << 6;
  const int lane = (int)threadIdx.x;
  const int lr = lane & 15;
  const int kh = lane >> 4;

  v8f c00, c01, c02, c03, c10, c11, c12, c13;
#pragma unroll
  for (int r = 0; r < 8; r++) {
    c00[r] = 0.f; c01[r] = 0.f; c02[r] = 0.f; c03[r] = 0.f;
    c10[r] = 0.f; c11[r] = 0.f; c12[r] = 0.f; c13[r] = 0.f;
  }

  const int n00 = n0 + lr;
  const int kmain = K & ~3;
  const int mr0 = m0 + lr, mr1 = m0 + 16 + lr;
  const bool mok0 = mr0 < M, mok1 = mr1 < M;
  const float* xrow0 = X + (size_t)(mok0 ? mr0 : 0) * ldx;
  const float* xrow1 = X + (size_t)(mok1 ? mr1 : 0) * ldx;
  const bool fast = (m0 + 32 <= M) && (n0 + 64 <= N);

  if (fast) {
    // -------- interior tile: unconditional loads, pointer stepping --------
    const float* xp0 = xrow0 + (kh << 1);
    const float* xp1 = xrow1 + (kh << 1);
    const float* wp = W + (size_t)(kh << 1) * ldw + n00;
    const size_t wstep = (size_t)ldw * 4;
    for (int kk = 0; kk < kmain; kk += 4) {
      v2f a0, a1;
      a0.x = xp0[0]; a0.y = xp0[1];
      a1.x = xp1[0]; a1.y = xp1[1];
      v2f b0, b1, b2, b3;
      b0.x = wp[0];  b0.y = wp[ldw];
      b1.x = wp[16]; b1.y = wp[ldw + 16];
      b2.x = wp[32]; b2.y = wp[ldw + 32];
      b3.x = wp[48]; b3.y = wp[ldw + 48];
      c00 = wmma4(a0, b0, c00);
      c10 = wmma4(a1, b0, c10);
      c01 = wmma4(a0, b1, c01);
      c11 = wmma4(a1, b1, c11);
      c02 = wmma4(a0, b2, c02);
      c12 = wmma4(a1, b2, c12);
      c03 = wmma4(a0, b3, c03);
      c13 = wmma4(a1, b3, c13);
      xp0 += 4; xp1 += 4; wp += wstep;
    }
  } else {
    // -------- N/M edge tile: clamp + select (rows ka in-bounds here) --------
    const int nc0 = (n00      < N) ? n00      : N - 1;
    const int nc1 = (n00 + 16 < N) ? n00 + 16 : N - 1;
    const int nc2 = (n00 + 32 < N) ? n00 + 32 : N - 1;
    const int nc3 = (n00 + 48 < N) ? n00 + 48 : N - 1;
    const bool nv0 = n00 < N, nv1 = n00 + 16 < N, nv2 = n00 + 32 < N, nv3 = n00 + 48 < N;
    for (int kk = 0; kk < kmain; kk += 4) {
      const int ka = kk + (kh << 1);
      float t00 = xrow0[ka], t01 = xrow0[ka + 1];
      float t10 = xrow1[ka], t11 = xrow1[ka + 1];
      v2f a0, a1;
      a0.x = mok0 ? t00 : 0.f;
      a0.y = mok0 ? t01 : 0.f;
      a1.x = mok1 ? t10 : 0.f;
      a1.y = mok1 ? t11 : 0.f;
      const float* w0 = W + (size_t)ka * ldw;
      const float* w1 = w0 + ldw;
      float u00 = w0[nc0], u10 = w1[nc0];
      float u01 = w0[nc1], u11 = w1[nc1];
      float u02 = w0[nc2], u12 = w1[nc2];
      float u03 = w0[nc3], u13 = w1[nc3];
      v2f b0, b1, b2, b3;
      b0.x = nv0 ? u00 : 0.f; b0.y = nv0 ? u10 : 0.f;
      b1.x = nv1 ? u01 : 0.f; b1.y = nv1 ? u11 : 0.f;
      b2.x = nv2 ? u02 : 0.f; b2.y = nv2 ? u12 : 0.f;
      b3.x = nv3 ? u03 : 0.f; b3.y = nv3 ? u13 : 0.f;
      c00 = wmma4(a0, b0, c00);
      c10 = wmma4(a1, b0, c10);
      c01 = wmma4(a0, b1, c01);
      c11 = wmma4(a1, b1, c11);
      c02 = wmma4(a0, b2, c02);
      c12 = wmma4(a1, b2, c12);
      c03 = wmma4(a0, b3, c03);
      c13 = wmma4(a1, b3, c13);
    }
  }

  if (kmain < K) {
    // -------- K tail (at most one step): clamp + select on K, N, M --------
    const int ka = kmain + (kh << 1);
    const int kc0 = (ka     < K) ? ka     : K - 1;
    const int kc1 = (ka + 1 < K) ? ka + 1 : K - 1;
    const bool kv0 = ka < K, kv1 = ka + 1 < K;
    float t00 = xrow0[kc0], t01 = xrow0[kc1];
    float t10 = xrow1[kc0], t11 = xrow1[kc1];
    v2f a0, a1;
    a0.x = (mok0 && kv0) ? t00 : 0.f;
    a0.y = (mok0 && kv1) ? t01 : 0.f;
    a1.x = (mok1 && kv0) ? t10 : 0.f;
    a1.y = (mok1 && kv1) ? t11 : 0.f;
    const float* w0 = W + (size_t)kc0 * ldw;
    const float* w1 = W + (size_t)kc1 * ldw;
    const bool nv0 = n00 < N, nv1 = n00 + 16 < N, nv2 = n00 + 32 < N, nv3 = n00 + 48 < N;
    const int nc0 = nv0 ? n00      : N - 1;
    const int nc1 = nv1 ? n00 + 16 : N - 1;
    const int nc2 = nv2 ? n00 + 32 : N - 1;
    const int nc3 = nv3 ? n00 + 48 : N - 1;
    float u00 = w0[nc0], u10 = w1[nc0];
    float u01 = w0[nc1], u11 = w1[nc1];
    float u02 = w0[nc2], u12 = w1[nc2];
    float u03 = w0[nc3], u13 = w1[nc3];
    v2f b0, b1, b2, b3;
    b0.x = (kv0 && nv0) ? u00 : 0.f; b0.y = (kv1 && nv0) ? u10 : 0.f;
    b1.x = (kv0 && nv1) ? u01 : 0.f; b1.y = (kv1 && nv1) ? u11 : 0.f;
    b2.x = (kv0 && nv2) ? u02 : 0.f; b2.y = (kv1 && nv2) ? u12 : 0.f;
    b3.x = (kv0 && nv3) ? u03 : 0.f; b3.y = (kv1 && nv3) ? u13 : 0.f;
    c00 = wmma4(a0, b0, c00);
    c10 = wmma4(a1, b0, c10);
    c01 = wmma4(a0, b1, c01);
    c11 = wmma4(a1, b1, c11);
    c02 = wmma4(a0, b2, c02);
    c12 = wmma4(a1, b2, c12);
    c03 = wmma4(a0, b3, c03);
    c13 = wmma4(a1, b3, c13);
  }

#if !USE_WMMA
  // Non-WMMA builds (host pass): recompute the full tile correctly (scalar).
  {
    v8f* accs[8] = {&c00, &c01, &c02, &c03, &c10, &c11, &c12, &c13};
    for (int h = 0; h < 2; h++)
      for (int t = 0; t < 4; t++) {
        const int n = n0 + (t << 4) + lr;
        for (int r = 0; r < 8; r++) {
          const int m = m0 + (h << 4) + (kh << 3) + r;
          float s = 0.f;
          if (m < M && n < N)
            for (int k = 0; k < K; k++) s += X[(size_t)m * ldx + k] * W[(size_t)k * ldw + n];
          (*accs[h * 4 + t])[r] = s;
        }
      }
  }
#endif

  // -------- store --------
  {
    const v8f* accs[8] = {&c00, &c01, &c02, &c03, &c10, &c11, &c12, &c13};
#pragma unroll
    for (int h = 0; h < 2; h++) {
#pragma unroll
      for (int t = 0; t < 4; t++) {
        const v8f& acc = *accs[h * 4 + t];
        const int n = n0 + (t << 4) + lr;
        if (n < N) {
          const float bs = bias ? bias[n] : 0.0f;
#pragma unroll
          for (int r = 0; r < 8; r++) {
            const int m = m0 + (h << 4) + (kh << 3) + r;
            if (m < M) {
              float v = acc[r] + bs;
              if (add) v += add[(size_t)m * ldadd + n];
              v = v > 0.0f ? v : v * slope;
              Y[(size_t)m * ldy + n] = v;
            }
          }
        }
      }
    }
  }
}

// ---------------- elementwise: out = act(a (+ b)) ----------------
__global__ void k_elt(const float* a, const float* b, float* out, size_t n, float slope) {
  size_t i = (size_t)blockIdx.x * blockDim.x + threadIdx.x;
  size_t st = (size_t)gridDim.x * blockDim.x;
  for (; i < n; i += st) {
    float v = a[i] + (b ? b[i] : 0.0f);
    out[i] = v > 0.0f ? v : v * slope;
  }
}

// ---------------- strided sub-matrix copy ----------------
__global__ void k_copy(const float* src, int lds_, float* dst, int ldd, int rows, int cols) {
  int i = blockIdx.x * blockDim.x + threadIdx.x;
  if (i >= rows * cols) return;
  int r = i / cols, c = i % cols;
  dst[(size_t)r * ldd + c] = src[(size_t)r * lds_ + c];
}

// ---------------- deterministic per-graph CSR build ----------------
// edges of graph g occupy e in [g*epg, (g+1)*epg); one thread per dst node.
__global__ void k_csr_count(const int* dst, const int* mask, int* cnt, int npg, int epg) {
  int g = blockIdx.x, t = threadIdx.x;
  if (t >= npg) return;
  int mynode = g * npg + t, c = 0, base = g * epg;
  for (int j = 0; j < epg; j++) {
    int e = base + j;
    if ((!mask || mask[e]) && dst[e] == mynode) c++;
  }
  cnt[mynode] = c;
}
__global__ void k_csr_fill(const int* dst, const int* mask, const int* rs, int* eid,
                           int npg, int epg) {
  int g = blockIdx.x, t = threadIdx.x;
  if (t >= npg) return;
  int mynode = g * npg + t, base = g * epg;
  int pos = rs[mynode];
  for (int j = 0; j < epg; j++) {
    int e = base + j;
    if ((!mask || mask[e]) && dst[e] == mynode) eid[pos++] = e;
  }
}
// 3-phase scan (n <= 65536)
__global__ void k_scan1(const int* in, int* incl, int* bsum, int n) {
  __shared__ int sh[256];
  int t = threadIdx.x, i = blockIdx.x * 256 + t;
  sh[t] = (i < n) ? in[i] : 0;
  __syncthreads();
  for (int o = 1; o < 256; o <<= 1) {
    int v = (t >= o) ? sh[t - o] : 0;
    __syncthreads();
    sh[t] += v;
    __syncthreads();
  }
  if (i < n) incl[i] = sh[t];
  if (t == 255) bsum[blockIdx.x] = sh[255];
}
__global__ void k_scan2(int* bsum, int nb) {
  __shared__ int sh[256];
  int t = threadIdx.x;
  sh[t] = (t < nb) ? bsum[t] : 0;
  __syncthreads();
  for (int o = 1; o < 256; o <<= 1) {
    int v = (t >= o) ? sh[t - o] : 0;
    __syncthreads();
    sh[t] += v;
    __syncthreads();
  }
  if (t < nb) bsum[t] = (t == 0) ? 0 : sh[t - 1];   // exclusive block offsets
}
__global__ void k_scan3(const int* incl, const int* bexcl, int* rs, int n) {
  int i = blockIdx.x * 256 + threadIdx.x;
  if (i < n) rs[i + 1] = incl[i] + bexcl[blockIdx.x];
  if (i == 0) rs[0] = 0;
}

// ---------------- self-loop edge attr: per-dst mean of incoming ea ----------------
__global__ void k_loop_attr(const float* ea, const int* rs, const int* eid, float* la, int n) {
  int i = blockIdx.x * blockDim.x + threadIdx.x;
  if (i >= n) return;
  float s[11];
#pragma unroll
  for (int k = 0; k < 11; k++) s[k] = 0.0f;
  int b = rs[i], e = rs[i + 1];
  for (int j = b; j < e; j++) {
    const float* p = ea + (size_t)eid[j] * 11;
#pragma unroll
    for (int k = 0; k < 11; k++) s[k] += p[k];
  }
  float c = (e - b) > 0 ? (float)(e - b) : 1.0f;
#pragma unroll
  for (int k = 0; k < 11; k++) la[(size_t)i * 11 + k] = s[k] / c;
}

// ---------------- GATv2 attention logits ----------------
// wave per edge (or per node when src==nullptr, for self-loops):
// out[e] = sum_d att[d] * lrelu( xl[s,d] + xr[t,d] + sum_k ea[e,k]*We[k,d], 0.2 )
__global__ void k_scores(const float* xl, const float* xr, const float* ea,
                         const int* src, const int* dst,
                         const float* We, const float* att,
                         float* out, int num, int D, int ldx) {
  extern __shared__ float sh[];
  float* sWe = sh;
  float* satt = sh + 11 * D;
  for (int i = threadIdx.x; i < 11 * D; i += blockDim.x) sWe[i] = We[i];
  for (int i = threadIdx.x; i < D; i += blockDim.x) satt[i] = att[i];
  __syncthreads();
  int lane = threadIdx.x & 31;
  int wid = (int)((blockIdx.x * blockDim.x + threadIdx.x) >> 5);
  int nw = (int)((gridDim.x * blockDim.x) >> 5);
  for (int e = wid; e < num; e += nw) {
    int s, t;
    if (src) { s = src[e]; t = dst[e]; } else { s = e; t = e; }
    const float* eap = ea + (size_t)e * 11;
    float ec[11];
#pragma unroll
    for (int k = 0; k < 11; k++) ec[k] = eap[k];
    float part = 0.0f;
    for (int d = lane; d < D; d += 32) {
      float v = xl[(size_t)s * ldx + d] + xr[(size_t)t * ldx + d];
#pragma unroll
      for (int k = 0; k < 11; k++) v += ec[k] * sWe[k * D + d];
      v = v > 0.0f ? v : 0.2f * v;
      part += satt[d] * v;
    }
    part = warp_sum32(part);
    if (lane == 0) out[e] = part;
  }
}

// ---------------- segment softmax stats (max + denom) ----------------
__global__ void k_stats(const int* rs, const int* eid, const float* esc, const float* lsc,
                        float* emax, float* denom, int n) {
  int i = blockIdx.x * blockDim.x + threadIdx.x;
  if (i >= n) return;
  int b = rs[i], e = rs[i + 1];
  float m = -1e30f;
  for (int j = b; j < e; j++) m = fmaxf(m, esc[eid[j]]);
  if (lsc) m = fmaxf(m, lsc[i]);
  if (m <= -1e30f) m = 0.0f;   // empty-segment guard (matches reference)
  float dn = 0.0f;
  for (int j = b; j < e; j++) dn += expf(esc[eid[j]] - m);
  if (lsc) dn += expf(lsc[i] - m);
  emax[i] = m;
  denom[i] = dn + 1e-16f;
}

// ---------------- attention-weighted aggregation ----------------
// wave per dst node: out[n,:] = sum_e alpha_e * xl[src_e,:] (+self) + bias
__global__ void k_aggregate(const float* xl, int ldx, const int* srcA,
                            const int* rs, const int* eid,
                            const float* esc, const float* lsc,
                            const float* emax, const float* denom,
                            const float* bias, float* out, int ldo, int n, int D) {
  int lane = threadIdx.x & 31;
  int wid = (int)((blockIdx.x * blockDim.x + threadIdx.x) >> 5);
  int nw = (int)((gridDim.x * blockDim.x) >> 5);
  const int C = D >> 5;  // D in {64,256,512} -> C in {2,8,16}
  for (int i = wid; i < n; i += nw) {
    float acc[16];
#pragma unroll
    for (int c = 0; c < 16; c++) acc[c] = 0.0f;
    const float m = emax[i], dn = denom[i];
    const int b = rs[i], e = rs[i + 1];
    for (int j = b; j < e; j++) {
      int ee = eid[j];
      float w = expf(esc[ee] - m) / dn;
      const float* row = xl + (size_t)srcA[ee] * ldx;
#pragma unroll
      for (int c = 0; c < 16; c++)
        if (c < C) acc[c] += w * row[lane + 32 * c];
    }
    if (lsc) {
      float w = expf(lsc[i] - m) / dn;
      const float* row = xl + (size_t)i * ldx;
#pragma unroll
      for (int c = 0; c < 16; c++)
        if (c < C) acc[c] += w * row[lane + 32 * c];
    }
#pragma unroll
    for (int c = 0; c < 16; c++)
      if (c < C) {
        int d = lane + 32 * c;
        out[(size_t)i * ldo + d] = acc[c] + bias[d];
      }
  }
}

// ---------------- ||w|| -> 1/norm ----------------
__global__ void k_norm(const float* w, float* out, int n) {
  __shared__ float sh[256];
  int t = threadIdx.x;
  float s = 0.0f;
  for (int i = t; i < n; i += 256) s += w[i] * w[i];
  sh[t] = s;
  __syncthreads();
  for (int o = 128; o > 0; o >>= 1) {
    if (t < o) sh[t] += sh[t + o];
    __syncthreads();
  }
  if (t == 0) out[0] = 1.0f / sqrtf(sh[0]);
}

// ---------------- TopK pooling scores ----------------
__global__ void k_poolscore(const float* h, int ldx, const float* pw, const float* invn,
                            float* sc, int n, int D) {
  int lane = threadIdx.x & 31;
  int wid = (int)((blockIdx.x * blockDim.x + threadIdx.x) >> 5);
  int nw = (int)((gridDim.x * blockDim.x) >> 5);
  for (int i = wid; i < n; i += nw) {
    float p = 0.0f;
    for (int d = lane; d < D; d += 32) p += h[(size_t)i * ldx + d] * pw[d];
    p = warp_sum32(p);
    if (lane == 0) sc[i] = tanhf(p * invn[0]);
  }
}

// ---------------- top-52-of-64 per graph (ties -> lower index) ----------------
__global__ void k_topk(const float* score, int* perm, int* newpos, int kkeep) {
  __shared__ float sv[64];
  __shared__ int taken[64];
  __shared__ float rv[64];
  __shared__ int ri[64];
  int g = blockIdx.x, t = threadIdx.x;
  int node = g * 64 + t;
  sv[t] = score[node];
  taken[t] = 0;
  newpos[node] = -1;
  __syncthreads();
  for (int r = 0; r < kkeep; r++) {
    rv[t] = taken[t] ? -3.0e38f : sv[t];
    ri[t] = t;
    __syncthreads();
    for (int off = 32; off > 0; off >>= 1) {
      if (t < off) {
        float a = rv[t], b = rv[t + off];
        int ia = ri[t], ib = ri[t + off];
        if (b > a || (b == a && ib < ia)) { rv[t] = b; ri[t] = ib; }
      }
      __syncthreads();
    }
    if (t == 0) {
      int w = ri[0];
      taken[w] = 1;
      perm[g * kkeep + r] = g * 64 + w;
      newpos[g * 64 + w] = g * kkeep + r;
    }
    __syncthreads();
  }
}

// ---------------- hp = h[perm] * score[perm] ----------------
__global__ void k_gather(const float* h, const int* perm, const float* sc, float* hp,
                         int rows, int D) {
  int idx = blockIdx.x * blockDim.x + threadIdx.x;
  if (idx >= rows * D) return;
  int i = idx / D, d = idx % D;
  int p = perm[i];
  hp[(size_t)i * D + d] = h[(size_t)p * D + d] * sc[p];
}

// ---------------- edge remap after pooling ----------------
__global__ void k_remap(const int* src, const int* dst, const int* newpos,
                        int* nsrc, int* ndst, int* mask, int E) {
  int e = blockIdx.x * blockDim.x + threadIdx.x;
  if (e >= E) return;
  int s = newpos[src[e]], t = newpos[dst[e]];
  mask[e] = (s >= 0 && t >= 0) ? 1 : 0;
  nsrc[e] = s >= 0 ? s : 0;
  ndst[e] = t >= 0 ? t : 0;
}

// ---------------- SAGE mean aggregation (D=256) ----------------
__global__ void k_sage_mean(const float* x, int ldx, const int* srcA,
                            const int* rs, const int* eid, float* out, int n) {
  int lane = threadIdx.x & 31;
  int wid = (int)((blockIdx.x * blockDim.x + threadIdx.x) >> 5);
  int nw = (int)((gridDim.x * blockDim.x) >> 5);
  for (int i = wid; i < n; i += nw) {
    float acc[8];
#pragma unroll
    for (int c = 0; c < 8; c++) acc[c] = 0.0f;
    int b = rs[i], e = rs[i + 1];
    for (int j = b; j < e; j++) {
      const float* row = x + (size_t)srcA[eid[j]] * ldx;
#pragma unroll
      for (int c = 0; c < 8; c++) acc[c] += row[lane + 32 * c];
    }
    float c0 = (e - b) > 0 ? (float)(e - b) : 1.0f;
#pragma unroll
    for (int c = 0; c < 8; c++) out[(size_t)i * 256 + lane + 32 * c] = acc[c] / c0;
  }
}

// ---------------- global mean+max pool (52 nodes per graph) ----------------
__global__ void k_gpool2(const float* x, int ldx, float* out, int D, int npg, int ng) {
  int idx = blockIdx.x * blockDim.x + threadIdx.x;
  if (idx >= ng * D) return;
  int g = idx / D, d = idx % D;
  const float* p = x + (size_t)g * npg * ldx + d;
  float s = 0.0f, m = -3.0e38f;
  for (int i = 0; i < npg; i++) {
    float v = p[(size_t)i * ldx];
    s += v;
    m = fmaxf(m, v);
  }
  out[(size_t)g * (2 * D) + d] = s / (float)npg;
  out[(size_t)g * (2 * D) + D + d] = m;
}

// ---------------- GCN adjacency normalization ----------------
__global__ void k_rowsum(const float* adj, float* dinv, int L) {
  int i = blockIdx.x * blockDim.x + threadIdx.x;
  if (i >= L) return;
  float s = 1.0f;  // +I diagonal
  for (int j = 0; j < L; j++) s += adj[(size_t)i * L + j];
  dinv[i] = 1.0f / sqrtf(s);
}
__global__ void k_normadj(const float* adj, const float* dinv, float* an, int L) {
  int idx = blockIdx.x * blockDim.x + threadIdx.x;
  if (idx >= L * L) return;
  int i = idx / L, j = idx % L;
  float a = adj[idx] + (i == j ? 1.0f : 0.0f);
  an[idx] = dinv[i] * a * dinv[j];
}

// =====================================================================
// host orchestration
// =====================================================================
extern "C" void kernel_launch(void* const* d_in, const int* in_sizes, int n_in,
                              void* d_out, int out_size, void* d_ws, size_t ws_size,
                              hipStream_t stream) {
  (void)in_sizes; (void)n_in; (void)out_size; (void)ws_size;

  const int NG = 1024, NPG = 64, EPG = 512, KKEEP = 52;
  const int NN = NG * NPG;    // 65536 nodes
  const int NP = NG * KKEEP;  // 53248 pooled nodes
  const int E = NG * EPG;     // 524288 edges

  auto F = [&](int i) { return (const float*)d_in[i]; };
  auto I = [&](int i) { return (const int*)d_in[i]; };

  // param index map (recursive insertion-order flattening of setup_inputs()):
  // 0:x 1:edge_attr 2:adj_coarse 3:adj_fine
  // 4..10:g1{Wl,bl,Wr,br,We,att,b} 11..17:g2 18:pool_w 19..21:sage{Wl,bl,Wr}
  // 22..28:bc.g1 29..35:bc.g2 36..41:bc{l1W,l1b,l2W,l2b,l3W,l3b}
  // 42..48:bf.g1 49..55:bf.g2 56..59:bf{l1W,l1b,l2W,l2b}
  // 60..66:b2.g1 67..73:b2.g2 74..77:b2{l1W,l1b,l2W,l2b}
  // 78:gcn1W 79:gcn1b 80:gcn2W 81:gcn2b 82:edge_index 83:batch
  const float* EA = F(1);
  const int* srcp = I(82);
  const int* dstp = srcp + E;

  // ---- workspace bump allocator ----
  char* wsb = (char*)d_ws;
  size_t off = 0;
  auto alloc = [&](size_t bytes) -> void* {
    void* p = (void*)(wsb + off);
    off += (bytes + 255) & ~(size_t)255;
    return p;
  };
  const size_t SLOT = (size_t)NP * 512;  // covers NN*256 too
  float* SA = (float*)alloc(SLOT * 4);
  float* SB = (float*)alloc(SLOT * 4);
  float* SC = (float*)alloc(SLOT * 4);
  float* HS = (float*)alloc((size_t)NP * 256 * 4);
  float* LA = (float*)alloc((size_t)NN * 11 * 4);
  float* ESC = (float*)alloc((size_t)E * 4);
  float* LSC = (float*)alloc((size_t)NN * 4);
  float* EMAX = (float*)alloc((size_t)NN * 4);
  float* DEN = (float*)alloc((size_t)NN * 4);
  float* SCOREP = (float*)alloc((size_t)NN * 4);
  int* PERM = (int*)alloc((size_t)NP * 4);
  int* NEWPOS = (int*)alloc((size_t)NN * 4);
  int* NSRC = (int*)alloc((size_t)E * 4);
  int* NDST = (int*)alloc((size_t)E * 4);
  int* EMSK = (int*)alloc((size_t)E * 4);   // (MASK2 is a HIP header macro)
  int* CNT = (int*)alloc((size_t)NN * 4);
  int* INCL = (int*)alloc((size_t)NN * 4);
  int* BSUM = (int*)alloc(256 * 4);
  int* RS0 = (int*)alloc((size_t)(NN + 1) * 4);
  int* RS1 = (int*)alloc((size_t)(NP + 1) * 4);
  int* EID0 = (int*)alloc((size_t)E * 4);
  int* EID1 = (int*)alloc((size_t)E * 4);
  float* X1 = (float*)alloc((size_t)1024 * 1024 * 4);
  float* X2 = (float*)alloc((size_t)1024 * 1024 * 4);
  float* XS = (float*)alloc((size_t)1024 * 1024 * 4);
  float* CAT1 = (float*)alloc((size_t)NG * 181 * 4);
  float* CAT2 = (float*)alloc((size_t)NG * 877 * 4);
  float* AN = (float*)alloc((size_t)877 * 877 * 4);
  float* DINV = (float*)alloc(877 * 4);
  float* TG = (float*)alloc((size_t)NG * 877 * 4);
  float* INVN = (float*)alloc(64);

  // ---- launchers ----
  auto gemmL = [&](const float* X, int ldx, const float* W, int ldw, const float* bias,
                   const float* add, int ldadd, float* Y, int ldy,
                   int M, int N, int K, float slope) {
    dim3 g((M + 31) / 32, (N + 63) / 64);
    k_gemm<<<g, 32, 0, stream>>>(X, ldx, W, ldw, bias, add, ldadd, Y, ldy, M, N, K, slope);
  };
  auto elt = [&](const float* a, const float* b, float* o, size_t n, float slope) {
    int blk = (int)((n + 255) / 256);
    k_elt<<<blk, 256, 0, stream>>>(a, b, o, n, slope);
  };
  auto scan = [&](const int* cnt, int n, int* rs) {
    int nb = (n + 255) / 256;
    k_scan1<<<nb, 256, 0, stream>>>(cnt, INCL, BSUM, n);
    k_scan2<<<1, 256, 0, stream>>>(BSUM, nb);
    k_scan3<<<nb, 256, 0, stream>>>(INCL, BSUM, rs, n);
  };
  auto run_gat = [&](const float* xin, int ldin, int Din, int Dout, int P,
                     const int* sA, const int* dA, const int* rs, const int* eid,
                     int nNodes, const float* la, float* xl, float* xr, float* out) {
    gemmL(xin, ldin, F(P + 0), Dout, F(P + 1), nullptr, 0, xl, Dout, nNodes, Dout, Din, 1.0f);
    gemmL(xin, ldin, F(P + 2), Dout, F(P + 3), nullptr, 0, xr, Dout, nNodes, Dout, Din, 1.0f);
    size_t shm = (size_t)(12 * Dout) * sizeof(float);
    k_scores<<<2048, 256, shm, stream>>>(xl, xr, EA, sA, dA, F(P + 4), F(P + 5), ESC, E, Dout, Dout);
    if (la)
      k_scores<<<2048, 256, shm, stream>>>(xl, xr, la, nullptr, nullptr, F(P + 4), F(P + 5),
                                           LSC, nNodes, Dout, Dout);
    k_stats<<<(nNodes + 255) / 256, 256, 0, stream>>>(rs, eid, ESC, la ? LSC : nullptr,
                                                      EMAX, DEN, nNodes);
    k_aggregate<<<(nNodes + 7) / 8, 256, 0, stream>>>(xl, Dout, sA, rs, eid, ESC,
                                                      la ? LSC : nullptr, EMAX, DEN,
                                                      F(P + 6), out, Dout, nNodes, Dout);
  };

  // ================= Phase A: shared GNN on 65536 nodes =================
  // CSR over original edges (all unmasked)
  k_csr_count<<<NG, 64, 0, stream>>>(dstp, nullptr, CNT, NPG, EPG);
  scan(CNT, NN, RS0);
  k_csr_fill<<<NG, 64, 0, stream>>>(dstp, nullptr, RS0, EID0, NPG, EPG);

  // g1: 30 -> 64, no self loops;  h1 = lrelu(gat(x))
  run_gat(F(0), 30, 30, 64, 4, srcp, dstp, RS0, EID0, NN, nullptr, SA, SB, SB);
  elt(SB, nullptr, SB, (size_t)NN * 64, 0.01f);
  // g2: 64 -> 256, self loops (loop attr = per-dst mean of incoming ea)
  k_loop_attr<<<(NN + 255) / 256, 256, 0, stream>>>(EA, RS0, EID0, LA, NN);
  run_gat(SB, 64, 64, 256, 11, srcp, dstp, RS0, EID0, NN, LA, SA, SC, SC);
  elt(SC, nullptr, SC, (size_t)NN * 256, 0.01f);  // h (65536 x 256) in SC

  // ================= Phase B: TopK pooling =================
  k_norm<<<1, 256, 0, stream>>>(F(18), INVN, 256);
  k_poolscore<<<(NN + 7) / 8, 256, 0, stream>>>(SC, 256, F(18), INVN, SCOREP, NN, 256);
  k_topk<<<NG, 64, 0, stream>>>(SCOREP, PERM, NEWPOS, KKEEP);
  k_gather<<<(NP * 256 + 255) / 256, 256, 0, stream>>>(SC, PERM, SCOREP, HS, NP, 256);
  k_remap<<<(E + 255) / 256, 256, 0, stream>>>(srcp, dstp, NEWPOS, NSRC, NDST, EMSK, E);
  // pooled CSR
  k_csr_count<<<NG, 64, 0, stream>>>(NDST, EMSK, CNT, KKEEP, EPG);
  scan(CNT, NP, RS1);
  k_csr_fill<<<NG, 64, 0, stream>>>(NDST, EMSK, RS1, EID1, KKEEP, EPG);
  k_loop_attr<<<(NP + 255) / 256, 256, 0, stream>>>(EA, RS1, EID1, LA, NP);

  // ================= Phase C: SAGE residual =================
  k_sage_mean<<<(NP + 7) / 8, 256, 0, stream>>>(HS, 256, NSRC, RS1, EID1, SA, NP);
  gemmL(SA, 256, F(19), 256, F(20), nullptr, 0, SB, 256, NP, 256, 256, 1.0f);        // lin_l(mean)
  gemmL(HS, 256, F(21), 256, nullptr, SB, 256, SB, 256, NP, 256, 256, 1.0f);         // + lin_r(x)
  elt(HS, SB, HS, (size_t)NP * 256, 0.01f);                                          // hs

  // ================= Phase D: task branches =================
  auto run_branch = [&](int Pg1, int Pg2, int Dmid, bool residual,
                        int l1W, int l1b, int l1N, int l2W, int l2b, int l2N,
                        int l3W, int l3b, int l3N, float* zout, int ldz) {
    run_gat(HS, 256, 256, Dmid, Pg1, NSRC, NDST, RS1, EID1, NP, LA, SA, SB, SB);
    if (residual) elt(HS, SB, SB, (size_t)NP * Dmid, 0.01f);
    else          elt(SB, nullptr, SB, (size_t)NP * Dmid, 0.01f);
    k_gpool2<<<(NG * Dmid + 255) / 256, 256, 0, stream>>>(SB, Dmid, X1, Dmid, KKEEP, NG);
    run_gat(SB, Dmid, Dmid, Dmid, Pg2, NSRC, NDST, RS1, EID1, NP, LA, SA, SC, SC);
    elt(SB, SC, SC, (size_t)NP * Dmid, 0.01f);
    k_gpool2<<<(NG * Dmid + 255) / 256, 256, 0, stream>>>(SC, Dmid, X2, Dmid, KKEEP, NG);
    elt(X1, X2, XS, (size_t)NG * 2 * Dmid, 1.0f);                        // x1 + x2
    gemmL(XS, 2 * Dmid, F(l1W), l1N, F(l1b), nullptr, 0, X1, l1N, NG, l1N, 2 * Dmid, 0.01f);
    if (l3W >= 0) {
      gemmL(X1, l1N, F(l2W), l2N, F(l2b), nullptr, 0, X2, l2N, NG, l2N, l1N, 0.01f);
      gemmL(X2, l2N, F(l3W), l3N, F(l3b), nullptr, 0, zout, ldz, NG, l3N, l2N, 1.0f);
    } else {
      gemmL(X1, l1N, F(l2W), l2N, F(l2b), nullptr, 0, zout, ldz, NG, l2N, l1N, 1.0f);
    }
  };

  float* OUTC = (float*)d_out;                 // [1024,11]
  float* OUTF = OUTC + (size_t)NG * 11;        // [1024,170]
  float* OUTF2 = OUTF + (size_t)NG * 170;      // [1024,707]

  // coarse branch -> out_coarse, and into cat1 cols 170..180
  run_branch(22, 29, 256, true, 36, 37, 256, 38, 39, 64, 40, 41, 11, OUTC, 11);
  k_copy<<<(NG * 11 + 255) / 256, 256, 0, stream>>>(OUTC, 11, CAT1 + 170, 181, NG, 11);
  // fine branch -> cat1 cols 0..169
  run_branch(42, 49, 256, true, 56, 57, 256, 58, 59, 170, -1, -1, 0, CAT1, 181);
  // gcn1: out_fine = (cat1 @ An181) @ W + b
  k_rowsum<<<(181 + 63) / 64, 64, 0, stream>>>(F(2), DINV, 181);
  k_normadj<<<(181 * 181 + 255) / 256, 256, 0, stream>>>(F(2), DINV, AN, 181);
  gemmL(CAT1, 181, AN, 181, nullptr, nullptr, 0, TG, 181, NG, 181, 181, 1.0f);
  gemmL(TG, 181, F(78), 170, F(79), nullptr, 0, OUTF, 170, NG, 170, 181, 1.0f);
  k_copy<<<(NG * 170 + 255) / 256, 256, 0, stream>>>(OUTF, 170, CAT2 + 707, 877, NG, 170);
  // fine2 branch -> cat2 cols 0..706
  run_branch(60, 67, 512, false, 74, 75, 512, 76, 77, 707, -1, -1, 0, CAT2, 877);
  // gcn2: out_fine2 = (cat2 @ An877) @ W + b
  k_rowsum<<<(877 + 63) / 64, 64, 0, stream>>>(F(3), DINV, 877);
  k_normadj<<<(877 * 877 + 255) / 256, 256, 0, stream>>>(F(3), DINV, AN, 877);
  gemmL(CAT2, 877, AN, 877, nullptr, nullptr, 0, TG, 877, NG, 877, 877, 1.0f);
  gemmL(TG, 877, F(80), 707, F(81), nullptr, 0, OUTF2, 707, NG, 707, 877, 1.0f);
}